// GAT_51694226374713
// MI455X (gfx1250) — compile-verified
//
#include <hip/hip_runtime.h>

typedef __attribute__((ext_vector_type(2))) float v2f;
typedef __attribute__((ext_vector_type(8))) float v8f;

#define NEG_SLOPE 0.2f

__device__ __forceinline__ float lrelu(float x) { return x > 0.f ? x : NEG_SLOPE * x; }

// float atomic max via sign-aware integer atomics (valid with -inf init)
__device__ __forceinline__ void atomicMaxF(float* addr, float v) {
    if (v >= 0.f) atomicMax((int*)addr, __float_as_int(v));
    else atomicMin((unsigned int*)addr, (unsigned int)__float_as_int(v));
}

// CDNA5 async copy: one 16-byte chunk global -> LDS, tracked by ASYNCcnt
__device__ __forceinline__ void async_g2l_b128(unsigned lds_byte_addr, const float* gaddr) {
    unsigned long long ga = (unsigned long long)gaddr;
    asm volatile("global_load_async_to_lds_b128 %0, %1, off"
                 :: "v"(lds_byte_addr), "v"(ga) : "memory");
}
__device__ __forceinline__ void wait_async0() {
    asm volatile("s_wait_asynccnt 0x0" ::: "memory");
}

__global__ void fill_f32(float* p, float v, int n) {
    int i = blockIdx.x * blockDim.x + threadIdx.x;
    if (i < n) p[i] = v;
}

// Wt[n*256+k] = W[k*256+n]   (W is [256,256])
__global__ void transpose_w1(const float* __restrict__ W, float* __restrict__ Wt) {
    int i = blockIdx.x * blockDim.x + threadIdx.x;
    if (i >= 256 * 256) return;
    int k = i >> 8, n = i & 255;
    Wt[n * 256 + k] = W[i];
}

// Wt[n*256+k] = W[k*40+n]    (W is [256,40])
__global__ void transpose_w2(const float* __restrict__ W, float* __restrict__ Wt) {
    int i = blockIdx.x * blockDim.x + threadIdx.x;
    if (i >= 256 * 40) return;
    int k = i / 40, n = i % 40;
    Wt[n * 256 + k] = W[i];
}

// C[M,256] = A[M,256] x B[256,256] with Bt = B^T ([n][k], contiguous in k).
// blockDim = 256 (8 waves); wave w computes rows [bid*16,+16) x cols [w*32,+32).
// A tile (16x256, 16KB) staged in LDS via async-to-LDS; A frags read with ds_load_b64.
__global__ void gat_gemm1_wmma(const float* __restrict__ A,
                               const float* __restrict__ Bt,
                               float* __restrict__ C) {
    __shared__ float sA[16 * 256];
    const int tid = threadIdx.x;
    const int m0  = blockIdx.x * 16;
    {
        const unsigned lbase = (unsigned)(unsigned long long)(void*)sA;
        const float* g = A + (size_t)m0 * 256 + tid * 16;   // 16 floats per thread
        #pragma unroll
        for (int i = 0; i < 4; ++i)
            async_g2l_b128(lbase + (unsigned)(tid * 64 + i * 16), g + i * 4);
        wait_async0();
        __syncthreads();
    }
    const int wave = tid >> 5;
    const int lane = tid & 31;
    const int lr   = lane & 15;
    const int hi   = lane >> 4;      // 0 or 1
    const int klo  = hi * 2;         // K sub-offset held by this half-wave
    const int n0   = wave * 32;

    v8f c0 = {}; v8f c1 = {};
    for (int k0 = 0; k0 < 256; k0 += 4) {
        v2f a  = *(const v2f*)&sA[lr * 256 + k0 + klo];                      // ds_load_b64
        v2f b0 = *(const v2f*)&Bt[(size_t)(n0 + lr)      * 256 + k0 + klo]; // global b64
        v2f b1 = *(const v2f*)&Bt[(size_t)(n0 + 16 + lr) * 256 + k0 + klo];
        c0 = __builtin_amdgcn_wmma_f32_16x16x4_f32(false, a, false, b0, (short)0, c0, false, false);
        c1 = __builtin_amdgcn_wmma_f32_16x16x4_f32(false, a, false, b1, (short)0, c1, false, false);
    }
    float* crow = C + (size_t)(m0 + hi * 8) * 256 + n0 + lr;
    #pragma unroll
    for (int r = 0; r < 8; ++r) {
        crow[(size_t)r * 256]      = c0[r];
        crow[(size_t)r * 256 + 16] = c1[r];
    }
}

// C[M,40] = A[M,256] x B[256,40], with B2t = B^T padded view ([n][k], n<40 valid).
// blockDim = 96 (3 waves), wave w -> cols [w*16,+16), masked to 40.
__global__ void gat_gemm2_wmma(const float* __restrict__ A,
                               const float* __restrict__ B2t,
                               float* __restrict__ C) {
    __shared__ float sA[16 * 256];
    const int tid = threadIdx.x;
    const int m0  = blockIdx.x * 16;
    {
        const unsigned lbase = (unsigned)(unsigned long long)(void*)sA;
        const float* gb = A + (size_t)m0 * 256;
        for (int i = tid; i < 1024; i += 96)                 // 1024 x b128 = 16KB
            async_g2l_b128(lbase + (unsigned)(i * 16), gb + i * 4);
        wait_async0();
        __syncthreads();
    }
    const int wave = tid >> 5;
    const int lane = tid & 31;
    const int lr   = lane & 15;
    const int hi   = lane >> 4;
    const int klo  = hi * 2;
    const int col  = wave * 16 + lr;
    const bool ok  = col < 40;

    v8f c0 = {};
    for (int k0 = 0; k0 < 256; k0 += 4) {
        v2f a = *(const v2f*)&sA[lr * 256 + k0 + klo];
        v2f b = { 0.f, 0.f };
        if (ok) b = *(const v2f*)&B2t[(size_t)col * 256 + k0 + klo];
        c0 = __builtin_amdgcn_wmma_f32_16x16x4_f32(false, a, false, b, (short)0, c0, false, false);
    }
    if (ok) {
        float* crow = C + (size_t)(m0 + hi * 8) * 40 + col;
        #pragma unroll
        for (int r = 0; r < 8; ++r) crow[(size_t)r * 40] = c0[r];
    }
}

// per-(node,head) attention coefficients, layer 1: h [N,8,32]
__global__ void gat_alphas1(const float* __restrict__ h,
                            const float* __restrict__ aw_s,
                            const float* __restrict__ aw_d,
                            float* __restrict__ as, float* __restrict__ ad, int N) {
    int i = blockIdx.x * blockDim.x + threadIdx.x;   // i = n*8 + head
    if (i >= N * 8) return;
    int head = i & 7;
    const float* hp = h + (size_t)(i >> 3) * 256 + head * 32;
    const float* sw = aw_s + head * 32;
    const float* dw = aw_d + head * 32;
    float s = 0.f, d = 0.f;
    #pragma unroll 8
    for (int c = 0; c < 32; ++c) { float v = hp[c]; s += v * sw[c]; d += v * dw[c]; }
    as[i] = s; ad[i] = d;
}

// per-node attention coefficients, layer 2: h [N,40], 1 head
__global__ void gat_alphas2(const float* __restrict__ h,
                            const float* __restrict__ aw_s,
                            const float* __restrict__ aw_d,
                            float* __restrict__ as, float* __restrict__ ad, int N) {
    int n = blockIdx.x * blockDim.x + threadIdx.x;
    if (n >= N) return;
    const float* hp = h + (size_t)n * 40;
    float s = 0.f, d = 0.f;
    #pragma unroll 8
    for (int c = 0; c < 40; ++c) { float v = hp[c]; s += v * aw_s[c]; d += v * aw_d[c]; }
    as[n] = s; ad[n] = d;
}

// segment max of leaky-relu logits over dst; idx in [0, E+N): last N are self-loops
__global__ void gat_edge_max(const int* __restrict__ ei, int E, int N, int H,
                             const float* __restrict__ as, const float* __restrict__ ad,
                             float* __restrict__ m) {
    int e = blockIdx.x * blockDim.x + threadIdx.x;
    if (e >= E + N) return;
    int s = (e < E) ? ei[e]     : (e - E);
    int d = (e < E) ? ei[E + e] : (e - E);
    for (int h = 0; h < H; ++h)
        atomicMaxF(&m[d * H + h], lrelu(as[s * H + h] + ad[d * H + h]));
}

// segment sum of exp(logit - max) over dst
__global__ void gat_edge_den(const int* __restrict__ ei, int E, int N, int H,
                             const float* __restrict__ as, const float* __restrict__ ad,
                             const float* __restrict__ m, float* __restrict__ den) {
    int e = blockIdx.x * blockDim.x + threadIdx.x;
    if (e >= E + N) return;
    int s = (e < E) ? ei[e]     : (e - E);
    int d = (e < E) ? ei[E + e] : (e - E);
    for (int h = 0; h < H; ++h) {
        float l = lrelu(as[s * H + h] + ad[d * H + h]);
        atomicAdd(&den[d * H + h], __expf(l - m[d * H + h]));
    }
}

// layer-1 aggregation: one block (256 thr) per edge; j = head*32 + c
__global__ void gat_edge_agg1(const int* __restrict__ ei, int E, int N,
                              const float* __restrict__ as, const float* __restrict__ ad,
                              const float* __restrict__ m, const float* __restrict__ den,
                              const float* __restrict__ h1, float* __restrict__ agg) {
    int e = blockIdx.x;
    int s = (e < E) ? ei[e]     : (e - E);
    int d = (e < E) ? ei[E + e] : (e - E);
    int j = threadIdx.x;
    int head = j >> 5;
    float l = lrelu(as[s * 8 + head] + ad[d * 8 + head]);
    float alpha = __expf(l - m[d * 8 + head]) / den[d * 8 + head];
    atomicAdd(&agg[(size_t)d * 256 + j], alpha * h1[(size_t)s * 256 + j]);
}

// layer-2 aggregation: one block (64 thr, 40 active) per edge
__global__ void gat_edge_agg2(const int* __restrict__ ei, int E, int N,
                              const float* __restrict__ as, const float* __restrict__ ad,
                              const float* __restrict__ m, const float* __restrict__ den,
                              const float* __restrict__ h2, float* __restrict__ out) {
    int e = blockIdx.x;
    int s = (e < E) ? ei[e]     : (e - E);
    int d = (e < E) ? ei[E + e] : (e - E);
    int c = threadIdx.x;
    if (c >= 40) return;
    float l = lrelu(as[s] + ad[d]);
    float alpha = __expf(l - m[d]) / den[d];
    atomicAdd(&out[(size_t)d * 40 + c], alpha * h2[(size_t)s * 40 + c]);
}

__global__ void gat_elu_bias(float* __restrict__ agg, const float* __restrict__ b, int total) {
    int i = blockIdx.x * blockDim.x + threadIdx.x;
    if (i >= total) return;
    float v = agg[i] + b[i & 255];
    agg[i] = v > 0.f ? v : __expf(v) - 1.f;
}

__global__ void gat_bias2(float* __restrict__ out, const float* __restrict__ b, int total) {
    int i = blockIdx.x * blockDim.x + threadIdx.x;
    if (i >= total) return;
    out[i] += b[i % 40];
}

extern "C" void kernel_launch(void* const* d_in, const int* in_sizes, int n_in,
                              void* d_out, int out_size, void* d_ws, size_t ws_size,
                              hipStream_t stream) {
    const float* x    = (const float*)d_in[0];
    const int*   ei   = (const int*)d_in[1];
    const float* w1   = (const float*)d_in[2];
    const float* as1w = (const float*)d_in[3];
    const float* ad1w = (const float*)d_in[4];
    const float* b1   = (const float*)d_in[5];
    const float* w2   = (const float*)d_in[6];
    const float* as2w = (const float*)d_in[7];
    const float* ad2w = (const float*)d_in[8];
    const float* b2   = (const float*)d_in[9];
    float* out = (float*)d_out;

    const int N = in_sizes[0] / 256;   // 50000
    const int E = in_sizes[1] / 2;     // 800000
    const int T = E + N;               // edges + self-loops

    float* ws  = (float*)d_ws;
    float* h1  = ws;                                   // N*256
    float* agg = h1  + (size_t)N * 256;                // N*256 (becomes layer-1 output)
    float* as1 = agg + (size_t)N * 256;                // N*8
    float* ad1 = as1 + (size_t)N * 8;                  // N*8
    float* m1  = ad1 + (size_t)N * 8;                  // N*8
    float* dn1 = m1  + (size_t)N * 8;                  // N*8
    float* as2 = dn1 + (size_t)N * 8;                  // N
    float* ad2 = as2 + N;                              // N
    float* m2  = ad2 + N;                              // N
    float* dn2 = m2  + N;                              // N
    float* w1t = dn2 + N;                              // 256*256
    float* w2t = w1t + 256 * 256;                      // 40*256
    float* h2  = h1;                                   // h1 dead after layer-1 aggregation

    const float NINF = -__builtin_inff();

    // init scratch + weight transposes (deterministic every call)
    fill_f32<<<(N * 8 + 255) / 256, 256, 0, stream>>>(m1, NINF, N * 8);
    fill_f32<<<(N * 8 + 255) / 256, 256, 0, stream>>>(dn1, 0.f, N * 8);
    fill_f32<<<(N * 256 + 255) / 256, 256, 0, stream>>>(agg, 0.f, N * 256);
    fill_f32<<<(N + 255) / 256, 256, 0, stream>>>(m2, NINF, N);
    fill_f32<<<(N + 255) / 256, 256, 0, stream>>>(dn2, 0.f, N);
    fill_f32<<<(N * 40 + 255) / 256, 256, 0, stream>>>(out, 0.f, N * 40);
    transpose_w1<<<(256 * 256 + 255) / 256, 256, 0, stream>>>(w1, w1t);
    transpose_w2<<<(256 * 40 + 255) / 256, 256, 0, stream>>>(w2, w2t);

    // ---- layer 1 ----
    gat_gemm1_wmma<<<N / 16, 256, 0, stream>>>(x, w1t, h1);
    gat_alphas1<<<(N * 8 + 255) / 256, 256, 0, stream>>>(h1, as1w, ad1w, as1, ad1, N);
    gat_edge_max<<<(T + 255) / 256, 256, 0, stream>>>(ei, E, N, 8, as1, ad1, m1);
    gat_edge_den<<<(T + 255) / 256, 256, 0, stream>>>(ei, E, N, 8, as1, ad1, m1, dn1);
    gat_edge_agg1<<<T, 256, 0, stream>>>(ei, E, N, as1, ad1, m1, dn1, h1, agg);
    gat_elu_bias<<<(N * 256 + 255) / 256, 256, 0, stream>>>(agg, b1, N * 256);

    // ---- layer 2 ----
    gat_gemm2_wmma<<<N / 16, 96, 0, stream>>>(agg, w2t, h2);
    gat_alphas2<<<(N + 255) / 256, 256, 0, stream>>>(h2, as2w, ad2w, as2, ad2, N);
    gat_edge_max<<<(T + 255) / 256, 256, 0, stream>>>(ei, E, N, 1, as2, ad2, m2);
    gat_edge_den<<<(T + 255) / 256, 256, 0, stream>>>(ei, E, N, 1, as2, ad2, m2, dn2);
    gat_edge_agg2<<<T, 64, 0, stream>>>(ei, E, N, as2, ad2, m2, dn2, h2, out);
    gat_bias2<<<(N * 40 + 255) / 256, 256, 0, stream>>>(out, b2, N * 40);
}